// YoloV3Loss_18098992185756
// MI455X (gfx1250) — compile-verified
//
#include <hip/hip_runtime.h>
#include <math.h>

typedef __attribute__((ext_vector_type(2))) float v2f;
typedef __attribute__((ext_vector_type(8))) float v8f;

#define BATCH 32
#define NGT   32
#define NCLS  81
#define RESZ  416.0f

// ---------------------------------------------------------------------------
// Wave32 sum-reduction through the matrix pipe:
// A(16x4): lane m<16 -> A[m][0]=v, A[m][1]=0 ; lane m+16 -> A[m][2]=v, A[m][3]=0
// B(4x16) = ones  =>  D[m][n] = v_m + v_{m+16} for every column n.
// D layout: vgpr r of lanes 0-15 holds M=r, lanes 16-31 hold M=8+r.
// sum(d0..d7) gives rows 0-7 (lanes<16) / rows 8-15 (lanes>=16); xor-16 shfl joins.
// ---------------------------------------------------------------------------
static __device__ __forceinline__ float wave_sum_wmma(float v) {
  v2f a; a.x = v;    a.y = 0.0f;
  v2f b; b.x = 1.0f; b.y = 1.0f;
  v8f c = {0.f,0.f,0.f,0.f,0.f,0.f,0.f,0.f};
  v8f d = __builtin_amdgcn_wmma_f32_16x16x4_f32(false, a, false, b,
                                                (short)0, c, false, false);
  float s = d[0]+d[1]+d[2]+d[3]+d[4]+d[5]+d[6]+d[7];
  s += __shfl_xor(s, 16, 32);
  return s;   // all lanes: sum of the 32 lane inputs
}

static __device__ __forceinline__ float sigm(float x) {
  return 1.0f / (1.0f + __expf(-x));
}

static __device__ __forceinline__ float iou_corner(
    float ax1,float ay1,float ax2,float ay2,float aarea,
    float bx1,float by1,float bx2,float by2,float barea) {
  float lx = fmaxf(ax1,bx1), ly = fmaxf(ay1,by1);
  float rx = fminf(ax2,bx2), ry = fminf(ay2,by2);
  float w = fmaxf(rx-lx,0.f), h = fmaxf(ry-ly,0.f);
  float inter = w*h;
  return inter / (aarea + barea - inter);
}

// ---------------------------------------------------------------------------
__global__ void k0_init(unsigned* __restrict__ scalemax,
                        int* __restrict__ assign,
                        float* __restrict__ acc, int tot) {
  int i = blockIdx.x*blockDim.x + threadIdx.x;
  if (i < 3*BATCH*NGT) scalemax[i] = 0u;
  if (i < tot)         assign[i]   = -1;
  if (i < 8)           acc[i]      = 0.f;
}

// one thread per anchor location (b, y, x, a) of one scale
__global__ void k1_iou(const float* __restrict__ gt,
                       const unsigned char* __restrict__ mask,
                       const float* __restrict__ anch, int S,
                       unsigned* __restrict__ scalemax,   // [B][N] (this scale)
                       float* __restrict__ maxiou) {      // [B][S*S*3]
  int b   = blockIdx.y;
  int loc = blockIdx.x*blockDim.x + threadIdx.x;
  int SS3 = S*S*3;
  __shared__ float gx1[NGT], gy1[NGT], gx2[NGT], gy2[NGT], gar[NGT];
  __shared__ unsigned char gm[NGT];
  __shared__ unsigned smax[NGT];
  if (threadIdx.x < NGT) {
    int n = threadIdx.x;
    const float* g = gt + (b*NGT + n)*4;
    float x1=g[0]*(float)S, y1=g[1]*(float)S, x2=g[2]*(float)S, y2=g[3]*(float)S;
    gx1[n]=x1; gy1[n]=y1; gx2[n]=x2; gy2[n]=y2;
    gar[n]=(x2-x1)*(y2-y1);
    gm[n]=mask[b*NGT+n];
    smax[n]=0u;
  }
  __syncthreads();
  if (loc < SS3) {
    int a = loc % 3;
    int x = (loc/3) % S;
    int y = loc/(3*S);
    float aw = anch[2*a+0]*(float)S/RESZ;
    float ah = anch[2*a+1]*(float)S/RESZ;
    float cx = (float)x + 0.5f, cy = (float)y + 0.5f;
    float ax1=cx-aw*0.5f, ay1=cy-ah*0.5f, ax2=cx+aw*0.5f, ay2=cy+ah*0.5f;
    float aarea=(ax2-ax1)*(ay2-ay1);
    float vmaxValid = 0.f;
    for (int n = 0; n < NGT; ++n) {
      float io = iou_corner(ax1,ay1,ax2,ay2,aarea,
                            gx1[n],gy1[n],gx2[n],gy2[n],gar[n]);
      atomicMax(&smax[n], __float_as_uint(io));     // raw max (for best-scale)
      if (gm[n]) vmaxValid = fmaxf(vmaxValid, io);  // valid max (for ignore)
    }
    maxiou[(size_t)b*SS3 + loc] = vmaxValid;
  }
  __syncthreads();
  if (threadIdx.x < NGT)
    atomicMax(&scalemax[b*NGT + threadIdx.x], smax[threadIdx.x]);
}

// one thread per (b, n): pick best scale, cell, best anchor -> scatter winner n
__global__ void k2_assign(const float* __restrict__ gt,
                          const unsigned char* __restrict__ mask,
                          const unsigned* __restrict__ scalemax, // [3][B][N]
                          const float* __restrict__ a0,
                          const float* __restrict__ a1,
                          const float* __restrict__ a2,
                          int* __restrict__ as0, int* __restrict__ as1,
                          int* __restrict__ as2) {
  int b = blockIdx.x, n = threadIdx.x;
  if (!mask[b*NGT+n]) return;
  float m0 = __uint_as_float(scalemax[(0*BATCH+b)*NGT+n]);
  float m1 = __uint_as_float(scalemax[(1*BATCH+b)*NGT+n]);
  float m2 = __uint_as_float(scalemax[(2*BATCH+b)*NGT+n]);
  int best = 0; float bv = m0;                 // first-max argmax like jnp
  if (m1 > bv) { bv = m1; best = 1; }
  if (m2 > bv) { bv = m2; best = 2; }
  const int SZ[3] = {13,26,52};
  int S = SZ[best];
  const float* anch = (best==0)?a0:((best==1)?a1:a2);
  int* assign       = (best==0)?as0:((best==1)?as1:as2);
  const float* g = gt + (b*NGT+n)*4;
  float cxf = (g[0]+g[2])*0.5f*(float)S;
  float cyf = (g[1]+g[3])*0.5f*(float)S;
  int cx = min(max((int)floorf(cxf),0), S-1);
  int cy = min(max((int)floorf(cyf),0), S-1);
  float bx1=g[0]*(float)S, by1=g[1]*(float)S, bx2=g[2]*(float)S, by2=g[3]*(float)S;
  float barea=(bx2-bx1)*(by2-by1);
  int j = 0; float jb = -1.f;
  float acx=(float)cx+0.5f, acy=(float)cy+0.5f;
  for (int a = 0; a < 3; ++a) {
    float aw=anch[2*a]*(float)S/RESZ, ah=anch[2*a+1]*(float)S/RESZ;
    float ax1=acx-aw*0.5f, ay1=acy-ah*0.5f, ax2=acx+aw*0.5f, ay2=acy+ah*0.5f;
    float io = iou_corner(ax1,ay1,ax2,ay2,(ax2-ax1)*(ay2-ay1),
                          bx1,by1,bx2,by2,barea);
    if (io > jb) { jb = io; j = a; }
  }
  assign[((b*S+cy)*S+cx)*3 + j] = n;  // race == jnp duplicate-scatter semantics
}

// one thread per anchor location; WMMA wave reduction of the 5 loss partials
__global__ void k3_loss(const float* __restrict__ pred,
                        const float* __restrict__ gt,
                        const int* __restrict__ labels,
                        const float* __restrict__ anch,
                        const int* __restrict__ assign,
                        const float* __restrict__ maxiou,
                        int S, float* __restrict__ acc) {
  int b   = blockIdx.y;
  int loc = blockIdx.x*blockDim.x + threadIdx.x;
  int SS3 = S*S*3;
  float t_xy=0.f, t_wh=0.f, t_obj=0.f, t_no=0.f, t_cls=0.f;
  if (loc < SS3) {
    int a = loc % 3;
    int x = (loc/3) % S;
    int y = loc/(3*S);
    const float* p = pred + ((size_t)(b*S+y)*S + x)*(3*(5+NCLS)) + a*(5+NCLS);
    float sobj = sigm(p[4]);
    int   n  = assign[(size_t)b*SS3 + loc];
    float mi = maxiou[(size_t)b*SS3 + loc];
    if (n >= 0) {
      t_obj = -__logf(fmaxf(sobj, 1e-12f));
      const float* g = gt + (b*NGT+n)*4;
      float cxf=(g[0]+g[2])*0.5f*(float)S, cyf=(g[1]+g[3])*0.5f*(float)S;
      float tx = cxf - floorf(cxf), ty = cyf - floorf(cyf);
      float sx = sigm(p[0]), sy = sigm(p[1]);
      t_xy = 0.5f*((sx-tx)*(sx-tx) + (sy-ty)*(sy-ty));
      float sw=(g[2]-g[0])*(float)S, sh=(g[3]-g[1])*(float)S;
      float wgw=anch[2*a]*(float)S/RESZ, wgh=anch[2*a+1]*(float)S/RESZ;
      float dwx = p[2] - __logf(sw/wgw);
      float dwy = p[3] - __logf(sh/wgh);
      t_wh = 0.5f*(dwx*dwx + dwy*dwy);
      int ci = labels[b*NGT+n] + 1;
      ci = min(max(ci,0), NCLS-1);
      float cl = 0.f;
      for (int c = 0; c < NCLS; ++c) {
        float s = sigm(p[5+c]);
        cl += (c==ci) ? -__logf(fmaxf(s,1e-12f))
                      : -__logf(fmaxf(1.f-s,1e-12f));
      }
      t_cls = cl;
    } else {
      float ign = (mi < 0.5f) ? 1.f : 0.f;
      t_no = ign * (-__logf(fmaxf(1.f-sobj, 1e-12f)));
    }
  }
  // re-converged here: EXEC all-ones for WMMA (wave32)
  float r0 = wave_sum_wmma(t_xy);
  float r1 = wave_sum_wmma(t_wh);
  float r2 = wave_sum_wmma(t_obj);
  float r3 = wave_sum_wmma(t_no);
  float r4 = wave_sum_wmma(t_cls);
  if ((threadIdx.x & 31) == 0) {
    atomicAdd(&acc[0], r0);
    atomicAdd(&acc[1], r1);
    atomicAdd(&acc[2], r2);
    atomicAdd(&acc[3], r3);
    atomicAdd(&acc[4], r4);
  }
}

__global__ void k4_final(const float* __restrict__ acc, float* __restrict__ out) {
  out[0] = (5.f*acc[0] + 5.f*acc[1] + acc[2] + 0.5f*acc[3] + acc[4]) * (1.f/32.f);
}

// ---------------------------------------------------------------------------
extern "C" void kernel_launch(void* const* d_in, const int* in_sizes, int n_in,
                              void* d_out, int out_size, void* d_ws, size_t ws_size,
                              hipStream_t stream) {
  const float* pred[3] = { (const float*)d_in[0], (const float*)d_in[1],
                           (const float*)d_in[2] };
  const float*         gt     = (const float*)d_in[3];
  const int*           labels = (const int*)d_in[4];
  const unsigned char* mask   = (const unsigned char*)d_in[5];
  const float* anch[3] = { (const float*)d_in[6], (const float*)d_in[7],
                           (const float*)d_in[8] };
  float* out = (float*)d_out;

  const int SZ[3]   = {13, 26, 52};
  const int SS3[3]  = {13*13*3, 26*26*3, 52*52*3};
  const int BASE[3] = {0, BATCH*SS3[0], BATCH*(SS3[0]+SS3[1])};
  const int TOT     = BATCH*(SS3[0]+SS3[1]+SS3[2]);   // 340704

  char* ws = (char*)d_ws;
  unsigned* scalemax = (unsigned*)ws;                              // 3*B*N u32
  float*    maxiou   = (float*)(ws + 3*BATCH*NGT*4);               // TOT f32
  int*      assign   = (int*)  (ws + 3*BATCH*NGT*4 + (size_t)TOT*4);
  float*    acc      = (float*)(ws + 3*BATCH*NGT*4 + (size_t)TOT*8);

  k0_init<<<(TOT+255)/256, 256, 0, stream>>>(scalemax, assign, acc, TOT);

  for (int s = 0; s < 3; ++s) {
    dim3 grid((SS3[s]+255)/256, BATCH);
    k1_iou<<<grid, 256, 0, stream>>>(gt, mask, anch[s], SZ[s],
                                     scalemax + s*BATCH*NGT,
                                     maxiou + BASE[s]);
  }

  k2_assign<<<BATCH, NGT, 0, stream>>>(gt, mask, scalemax,
                                       anch[0], anch[1], anch[2],
                                       assign + BASE[0], assign + BASE[1],
                                       assign + BASE[2]);

  for (int s = 0; s < 3; ++s) {
    dim3 grid((SS3[s]+255)/256, BATCH);
    k3_loss<<<grid, 256, 0, stream>>>(pred[s], gt, labels, anch[s],
                                      assign + BASE[s], maxiou + BASE[s],
                                      SZ[s], acc);
  }

  k4_final<<<1, 1, 0, stream>>>(acc, out);
}